// HP_14216341750039
// MI455X (gfx1250) — compile-verified
//
#include <hip/hip_runtime.h>
#include <hip/hip_bf16.h>
#include <math.h>

// Hawkes-process NLL on MI455X (gfx1250, wave32).
//
// Algorithm: exponential-kernel scan  E[b,i,v] = sum_{j<i,t_j=v} w*exp(-w*(t_i-t_j))
// (O(B*L*K) instead of the reference's O(B*L^2) pair sum), then a dense f32 WMMA GEMM
// R = E (32736x64) x AhatT (64x64), gathering R[row, t_i] as the excitation term.
// All reductions use fixed-order trees (no float atomics) -> bitwise deterministic.

typedef __attribute__((ext_vector_type(2))) float v2f;
typedef __attribute__((ext_vector_type(8))) float v8f;

#define HP_EPS 1e-10f
#define HB 32
#define HL 1024
#define HK1 51          // K+1 event types
#define HKP 64          // padded type dimension (WMMA K and N)
#define HNR (HB*(HL-1)) // 32736 target rows (divisible by 16)
#define HNT (HNR/16)    // 2046 row tiles

// workspace layout (float offsets); total = (6400 + 32736*64)*4 B ~ 8.43 MB
#define OFF_AHT   0      // 64x64: AhatT[v*64+u] = softplus(alpha[u][v]), zero-padded
#define OFF_MHAT  4096   // 64: softplus(mu), zero-padded
#define OFF_ACOL  4160   // 64: a_colsum[k] = sum_{u=1..50} softplus(alpha[u][k])
#define OFF_OMEGA 4224   // softplus(beta)
#define OFF_SMH   4225   // sum(mhat[1:])
#define OFF_COMP  4240   // 32 per-batch compensators
#define OFF_LOGS  4288   // 2046 per-tile log-rate sums
#define OFF_E     6400   // HNR x 64 f32 scan output (row-major, 64-padded)

__device__ __forceinline__ float hp_sp(float x) {
    // numerically stable softplus, matches jax.nn.softplus
    return fmaxf(x, 0.0f) + log1pf(expf(-fabsf(x)));
}

// ---------------- kernel 1: parameter transforms ----------------
__global__ void hp_prep(const float* __restrict__ alpha,
                        const float* __restrict__ beta,
                        const float* __restrict__ mu,
                        float* __restrict__ ws) {
    int tid = threadIdx.x;
    for (int idx = tid; idx < HKP * HKP; idx += blockDim.x) {
        int v = idx >> 6, u = idx & 63;              // AhatT[v][u] = Ahat[u][v]
        float val = 0.0f;
        if (u < HK1 && v < HK1) val = hp_sp(alpha[u * HK1 + v]);
        ws[OFF_AHT + idx] = val;
    }
    for (int idx = tid; idx < HKP; idx += blockDim.x) {
        ws[OFF_MHAT + idx] = (idx < HK1) ? hp_sp(mu[idx]) : 0.0f;
        float ac = 0.0f;
        if (idx < HK1)
            for (int u = 1; u < HK1; ++u) ac += hp_sp(alpha[u * HK1 + idx]);
        ws[OFF_ACOL + idx] = ac;
    }
    if (tid == 0) {
        ws[OFF_OMEGA] = hp_sp(beta[0]);
        float s = 0.0f;
        for (int k = 1; k < HK1; ++k) s += hp_sp(mu[k]);
        ws[OFF_SMH] = s;
    }
}

// ---------------- kernel 2: per-batch decay scan (O(L*K)) ----------------
// block b = batch; thread v in [0,64) owns one (padded) type slot.
__global__ void hp_scan(const int* __restrict__ type_seq,
                        const float* __restrict__ time_seq,
                        float* __restrict__ ws) {
    int b = blockIdx.x;
    int v = threadIdx.x;
    float omega = ws[OFF_OMEGA];
    const int*   ty = type_seq + b * HL;
    const float* tm = time_seq + b * HL;
    float* out = ws + OFF_E + (size_t)b * (HL - 1) * HKP;
    float E = 0.0f;
    float tprev = tm[0];
    int   typrev = ty[0];
    for (int i = 1; i < HL; ++i) {
        float tcur = tm[i];
        float d = __expf(-omega * (tcur - tprev));     // dt >= 0 (cumsum), no overflow
        float add = (typrev != 0 && typrev == v) ? omega : 0.0f;  // pad (type 0) excluded
        E = d * (E + add);
        out[(size_t)(i - 1) * HKP + v] = E;
        tprev = tcur;
        typrev = ty[i];
    }
}

// ---------------- kernel 3: per-batch compensator ----------------
__global__ void hp_comp(const int* __restrict__ type_seq,
                        const float* __restrict__ time_seq,
                        float* __restrict__ ws) {
    __shared__ float red[256];
    int b = blockIdx.x, tid = threadIdx.x;
    float omega = ws[OFF_OMEGA];
    const int*   ty = type_seq + b * HL;
    const float* tm = time_seq + b * HL;
    float mx = -3.4e38f;
    for (int j = tid; j < HL; j += 256) mx = fmaxf(mx, tm[j]);
    red[tid] = mx; __syncthreads();
    for (int s = 128; s > 0; s >>= 1) {
        if (tid < s) red[tid] = fmaxf(red[tid], red[tid + s]);
        __syncthreads();
    }
    float maxt = red[0]; __syncthreads();
    float acc = 0.0f;
    for (int j = tid; j < HL; j += 256) {
        int t = ty[j];
        if (t != 0)
            acc += ws[OFF_ACOL + t] * (1.0f - __expf(-omega * (maxt - tm[j])));
    }
    red[tid] = acc; __syncthreads();
    for (int s = 128; s > 0; s >>= 1) {        // fixed tree -> deterministic
        if (tid < s) red[tid] += red[tid + s];
        __syncthreads();
    }
    if (tid == 0) ws[OFF_COMP + b] = maxt * ws[OFF_SMH] + red[0];
}

// ---------------- kernel 4: WMMA GEMM R = E x AhatT + gather + log ----------------
// one wave per 16-row tile; 8 waves / block; no barriers (per-wave LDS slices).
__global__ __launch_bounds__(256) void hp_gemm(const int* __restrict__ type_seq,
                                               float* __restrict__ ws) {
    __shared__ float xb_all[8 * 16 * HKP];   // 32 KB: per-wave 16x64 result tile
    int wave = threadIdx.x >> 5;
    int lane = threadIdx.x & 31;
    int tile = blockIdx.x * 8 + wave;
    if (tile >= HNT) return;
    float* xb = xb_all + wave * 16 * HKP;
    const float* Emat = ws + OFF_E;
    const float* Bt   = ws + OFF_AHT;        // B[k][n] = AhatT[k*64+n]

    int m   = lane & 15;       // row within tile (A operand)
    int kh  = lane >> 4;       // K half: lanes 0-15 -> k0,k0+1 ; lanes 16-31 -> k0+2,k0+3
    int col = lane & 15;       // column within N tile (B/C/D operand)
    int row0 = tile * 16;

    // A fragments: E[row0+m][k], k striped per ISA 16x4 f32 layout; 8B-aligned float2 loads
    v2f afr[16];
    const float* Erow = Emat + (size_t)(row0 + m) * HKP + 2 * kh;
    #pragma unroll
    for (int kk = 0; kk < 16; ++kk)
        afr[kk] = *(const v2f*)(Erow + 4 * kk);

    #pragma unroll
    for (int nt = 0; nt < 4; ++nt) {
        v8f acc = {};
        #pragma unroll
        for (int kk = 0; kk < 16; ++kk) {
            int k0 = 4 * kk + 2 * kh;
            v2f bfr;
            bfr.x = Bt[(k0    ) * HKP + nt * 16 + col];
            bfr.y = Bt[(k0 + 1) * HKP + nt * 16 + col];
            // 8 args: (neg_a, A, neg_b, B, c_mod, C, reuse_a, reuse_b)
            acc = __builtin_amdgcn_wmma_f32_16x16x4_f32(
                false, afr[kk], false, bfr, (short)0, acc, false, false);
        }
        // spill D tile to LDS: VGPR r holds rows r (lanes 0-15) / r+8 (lanes 16-31)
        int rowOff = kh * 8;
        #pragma unroll
        for (int r = 0; r < 8; ++r)
            xb[(rowOff + r) * HKP + nt * 16 + col] = acc[r];
    }
    asm volatile("s_wait_dscnt 0" ::: "memory");   // intra-wave LDS RAW fence (CDNA5 split counter)

    float lsum = 0.0f;
    if (lane < 16) {
        int flat = row0 + lane;                    // flat row -> (batch, target i)
        int b = flat / (HL - 1);
        int i = flat % (HL - 1) + 1;
        int t = type_seq[b * HL + i];
        if (t != 0) {                              // pad targets: rate=1 -> log=0
            float ag   = xb[lane * HKP + t];       // gather R[row, t_i]
            float rate = fmaxf(ws[OFF_MHAT + t] + ag, HP_EPS);
            lsum = __logf(rate);
        }
    }
    for (int off = 16; off > 0; off >>= 1)         // fixed-order wave reduction
        lsum += __shfl_down(lsum, off, 32);
    if (lane == 0) ws[OFF_LOGS + tile] = lsum;
}

// ---------------- kernel 5: deterministic finalize ----------------
__global__ void hp_final(const float* __restrict__ ws, float* __restrict__ out) {
    __shared__ float red[256];
    int tid = threadIdx.x;
    float s = 0.0f;
    for (int idx = tid; idx < HNT; idx += 256) s -= ws[OFF_LOGS + idx];
    for (int idx = tid; idx < HB;  idx += 256) s += ws[OFF_COMP + idx];
    red[tid] = s; __syncthreads();
    for (int st = 128; st > 0; st >>= 1) {
        if (tid < st) red[tid] += red[tid + st];
        __syncthreads();
    }
    if (tid == 0) out[0] = red[0];
}

extern "C" void kernel_launch(void* const* d_in, const int* in_sizes, int n_in,
                              void* d_out, int out_size, void* d_ws, size_t ws_size,
                              hipStream_t stream) {
    const int*   type_seq = (const int*)  d_in[0];
    const float* time_seq = (const float*)d_in[1];
    const float* alpha    = (const float*)d_in[2];
    const float* beta     = (const float*)d_in[3];
    const float* mu       = (const float*)d_in[4];
    float* ws = (float*)d_ws;   // requires ~8.5 MB scratch

    hp_prep <<<1, 256, 0, stream>>>(alpha, beta, mu, ws);
    hp_scan <<<HB, HKP, 0, stream>>>(type_seq, time_seq, ws);
    hp_comp <<<HB, 256, 0, stream>>>(type_seq, time_seq, ws);
    hp_gemm <<<(HNT + 7) / 8, 256, 0, stream>>>(type_seq, ws);
    hp_final<<<1, 256, 0, stream>>>(ws, (float*)d_out);
}